// TrajModel_46273977647341
// MI455X (gfx1250) — compile-verified
//
#include <hip/hip_runtime.h>
#include <math.h>

#define NXg 256
#define NYg 256
#define Mg  (32 * 513)   // 16416 non-uniform samples
#define Bg  4            // batch (channels)

typedef __attribute__((ext_vector_type(2))) float v2f;
typedef __attribute__((ext_vector_type(8))) float v8f;

// D = A(16x4 f32) * B(4x16 f32) + C(16x16 f32)  -- V_WMMA_F32_16X16X4_F32
__device__ __forceinline__ v8f wmma_f32_16x16x4(v2f a, v2f b, v8f c) {
  return __builtin_amdgcn_wmma_f32_16x16x4_f32(
      /*neg_a=*/false, a, /*neg_b=*/false, b,
      /*c_mod=*/(short)0, c, /*reuse_a=*/false, /*reuse_b=*/false);
}

#define V8F_ZERO {0.f,0.f,0.f,0.f,0.f,0.f,0.f,0.f}

// ---------------------------------------------------------------------------
// Tables.
//  m-paired (for type1 fragments; lane K = 2*half+j indexes consecutive m):
//    ExC[p*256+x] = (cos(kx_{2p} gx), cos(kx_{2p+1} gx)),  ExS = sin pairs
//    EyC/EyS likewise over y.
//  planar planes (for type2 fragments; lane K = y, v2f over adjacent y):
//    EycP[m*256+y] = cos(ky_m gy), EysP = sin.
// ---------------------------------------------------------------------------
__global__ void build_tables_kernel(const float* __restrict__ traj,
                                    v2f* __restrict__ ExC, v2f* __restrict__ ExS,
                                    v2f* __restrict__ EyC, v2f* __restrict__ EyS,
                                    float* __restrict__ EycP, float* __restrict__ EysP) {
  const int idx = blockIdx.x * blockDim.x + threadIdx.x;
  if (idx >= (Mg / 2) * NXg) return;
  const int p   = idx >> 8;
  const int col = idx & 255;
  const float g = (float)(col - (NXg / 2));
  float sx0, cx0, sx1, cx1, sy0, cy0, sy1, cy1;
  __sincosf(traj[2 * p] * g, &sx0, &cx0);
  __sincosf(traj[2 * p + 1] * g, &sx1, &cx1);
  __sincosf(traj[Mg + 2 * p] * g, &sy0, &cy0);
  __sincosf(traj[Mg + 2 * p + 1] * g, &sy1, &cy1);
  v2f t;
  t[0] = cx0; t[1] = cx1; ExC[idx] = t;
  t[0] = sx0; t[1] = sx1; ExS[idx] = t;
  t[0] = cy0; t[1] = cy1; EyC[idx] = t;
  t[0] = sy0; t[1] = sy1; EyS[idx] = t;
  EycP[(2 * p) * NYg + col] = cy0;  EycP[(2 * p + 1) * NYg + col] = cy1;
  EysP[(2 * p) * NYg + col] = sy0;  EysP[(2 * p + 1) * NYg + col] = sy1;
}

__global__ void init_w_kernel(float* __restrict__ w_r, float* __restrict__ w_i) {
  const int i = blockIdx.x * blockDim.x + threadIdx.x;
  if (i < Mg) { w_r[i] = 1.0f; w_i[i] = 0.0f; }
}

__global__ void zero_kernel(float* __restrict__ p, int n) {
  const int i = blockIdx.x * blockDim.x + threadIdx.x;
  if (i < n) p[i] = 0.0f;
}

// ---------------------------------------------------------------------------
// type1 (adjoint gridding): out[x,y] = sum_m coef[m]*conj(Ex[m,x])*conj(Ey[m,y])
// One block = 16(x) x 32(y) strip; A-fragment reused across 2 y-tiles.
// All fragment loads are b64 pairs already in WMMA lane order (no shuffles).
// ---------------------------------------------------------------------------
__global__ void __launch_bounds__(256) type1_kernel(
    const v2f* __restrict__ ExC, const v2f* __restrict__ ExS,
    const v2f* __restrict__ EyC, const v2f* __restrict__ EyS,
    const float* __restrict__ coef_r, const float* __restrict__ coef_i,
    float* __restrict__ out_r, float* __restrict__ out_i) {
  __shared__ float red_r[8][256];
  __shared__ float red_i[8][256];
  const int wave = threadIdx.x >> 5;
  const int lane = threadIdx.x & 31;
  const int half = lane >> 4;
  const int l16  = lane & 15;
  const int x0 = blockIdx.x * 16;
  const int y0 = blockIdx.y * 32;               // 2 y-tiles per block
  const int b  = blockIdx.z;
  // planar coef read as v2f == pair over consecutive m
  const v2f* __restrict__ cfR = (const v2f*)(coef_r + (size_t)b * Mg);
  const v2f* __restrict__ cfI = (const v2f*)(coef_i + (size_t)b * Mg);

  const int kchunk = Mg / 8;                    // 2052 (multiple of 4)
  const int kbeg = wave * kchunk;
  const int kend = kbeg + kchunk;

  v8f cr0 = V8F_ZERO, ci0 = V8F_ZERO;
  v8f cr1 = V8F_ZERO, ci1 = V8F_ZERO;

  for (int k = kbeg; k < kend; k += 4) {
    const int p = (k >> 1) + half;              // pair index: rows (2p, 2p+1)
    // A[x,m] = conj(Ex)*coef  (pairs over j=0,1 directly in lane order)
    const v2f eC = ExC[p * NXg + x0 + l16];
    const v2f eS = ExS[p * NXg + x0 + l16];
    const v2f wR = cfR[p];
    const v2f wI = cfI[p];
    const v2f ar  = eC * wR - eS * wI;
    const v2f ai  = eC * wI + eS * wR;
    const v2f nai = -ai;
    // B[m,y] = conj(Ey) = c + i s  (direct b64 pair loads)
    const v2f b0r = EyC[p * NYg + y0 + l16];
    const v2f b0i = EyS[p * NYg + y0 + l16];
    const v2f b1r = EyC[p * NYg + y0 + 16 + l16];
    const v2f b1i = EyS[p * NYg + y0 + 16 + l16];

    cr0 = wmma_f32_16x16x4(ar,  b0r, cr0);
    cr0 = wmma_f32_16x16x4(nai, b0i, cr0);
    ci0 = wmma_f32_16x16x4(ar,  b0i, ci0);
    ci0 = wmma_f32_16x16x4(ai,  b0r, ci0);
    cr1 = wmma_f32_16x16x4(ar,  b1r, cr1);
    cr1 = wmma_f32_16x16x4(nai, b1i, cr1);
    ci1 = wmma_f32_16x16x4(ar,  b1i, ci1);
    ci1 = wmma_f32_16x16x4(ai,  b1r, ci1);
  }

#pragma unroll
  for (int t = 0; t < 2; ++t) {                 // per y-tile LDS reduce
#pragma unroll
    for (int r = 0; r < 8; ++r) {
      const int e = (r + 8 * half) * 16 + l16;  // C/D: row = r+8*half, col = l16
      red_r[wave][e] = t ? cr1[r] : cr0[r];
      red_i[wave][e] = t ? ci1[r] : ci0[r];
    }
    __syncthreads();
    const int tt = threadIdx.x;
    float sr = 0.f, si = 0.f;
#pragma unroll
    for (int wv = 0; wv < 8; ++wv) { sr += red_r[wv][tt]; si += red_i[wv][tt]; }
    const int xx = x0 + (tt >> 4);
    const int yy = y0 + t * 16 + (tt & 15);
    const size_t o = (size_t)b * (NXg * NYg) + (size_t)xx * NYg + yy;
    out_r[o] = sr;
    out_i[o] = si;
    __syncthreads();
  }
}

// ---------------------------------------------------------------------------
// type2 (fused): s[m] += sum_x Ex[m,x] * (sum_y Ey[m,y] * in[x,y])
// A = Ey planar plane, v2f over adjacent y (lane order direct).
// B = input plane rows, v2f over adjacent y (lane order direct).
// One wave = one m-tile x 2 x-tiles; cross-lane reduce over x; atomics.
// ---------------------------------------------------------------------------
template <bool HAS_IMAG>
__global__ void __launch_bounds__(64) type2_kernel(
    const float* __restrict__ ExCf, const float* __restrict__ ExSf,  // m-paired (epilogue)
    const v2f* __restrict__ EycP, const v2f* __restrict__ EysP,      // planar planes
    const v2f* __restrict__ inR, const v2f* __restrict__ inI,        // planar r/i planes
    float* __restrict__ s) {          // interleaved (re,im), atomic accumulate
  const int wave = threadIdx.x >> 5;
  const int lane = threadIdx.x & 31;
  const int half = lane >> 4;
  const int l16  = lane & 15;
  const int m0 = (blockIdx.x * 2 + wave) * 16;  // m-tile
  const int x0 = blockIdx.y * 32;               // 2 x-tiles
  const int b  = blockIdx.z;
  const v2f* __restrict__ gR = inR + (size_t)b * (NXg * NYg / 2);
  const v2f* __restrict__ gI = HAS_IMAG ? (inI + (size_t)b * (NXg * NYg / 2)) : nullptr;

  v8f t0r = V8F_ZERO, t0i = V8F_ZERO;
  v8f t1r = V8F_ZERO, t1i = V8F_ZERO;

  const int arow = m0 + l16;
  const int brow0 = x0 + l16;
  const int brow1 = x0 + 16 + l16;

  for (int y = 0; y < NYg; y += 4) {
    const int yk0 = y + 2 * half;               // lane K pair (yk0, yk0+1)
    const v2f aC = EycP[(arow * NYg + yk0) >> 1];
    const v2f aS = EysP[(arow * NYg + yk0) >> 1];
    const v2f ar = aC;                          // Ey = c - i s
    const v2f ai = -aS;
    const v2f b0r = gR[(brow0 * NYg + yk0) >> 1];
    const v2f b1r = gR[(brow1 * NYg + yk0) >> 1];

    t0r = wmma_f32_16x16x4(ar, b0r, t0r);
    t0i = wmma_f32_16x16x4(ai, b0r, t0i);
    t1r = wmma_f32_16x16x4(ar, b1r, t1r);
    t1i = wmma_f32_16x16x4(ai, b1r, t1i);
    if (HAS_IMAG) {
      const v2f nai = aS;                       // -ai = +s
      const v2f b0i = gI[(brow0 * NYg + yk0) >> 1];
      const v2f b1i = gI[(brow1 * NYg + yk0) >> 1];
      t0r = wmma_f32_16x16x4(nai, b0i, t0r);
      t0i = wmma_f32_16x16x4(ar,  b0i, t0i);
      t1r = wmma_f32_16x16x4(nai, b1i, t1r);
      t1i = wmma_f32_16x16x4(ar,  b1i, t1i);
    }
  }

  // multiply by Ex[m,x] = exc - i*exs for both x-tiles, reduce over x
#pragma unroll
  for (int r = 0; r < 8; ++r) {
    const int m = m0 + r + 8 * half;
    const size_t base = 2 * (size_t)((m >> 1) * NXg) + (m & 1);  // m-paired table addr
    const float e0c = ExCf[base + 2 * (x0 + l16)];
    const float e0s = ExSf[base + 2 * (x0 + l16)];
    const float e1c = ExCf[base + 2 * (x0 + 16 + l16)];
    const float e1s = ExSf[base + 2 * (x0 + 16 + l16)];
    float pr = e0c * t0r[r] + e0s * t0i[r] + e1c * t1r[r] + e1s * t1i[r];
    float pi = e0c * t0i[r] - e0s * t0r[r] + e1c * t1i[r] - e1s * t1r[r];
#pragma unroll
    for (int off = 8; off >= 1; off >>= 1) {    // stays within 16-lane half
      pr += __shfl_xor(pr, off);
      pi += __shfl_xor(pi, off);
    }
    if (l16 == 0) {
      atomicAdd(&s[2 * ((size_t)b * Mg + m) + 0], pr);
      atomicAdd(&s[2 * ((size_t)b * Mg + m) + 1], pi);
    }
  }
}

__global__ void update_w_kernel(float* __restrict__ w_r, float* __restrict__ w_i,
                                const float* __restrict__ s) {
  const int i = blockIdx.x * blockDim.x + threadIdx.x;
  if (i >= Mg) return;
  const float sr = s[2 * i], si = s[2 * i + 1];
  const float inv = 1.0f / sqrtf(sr * sr + si * si);
  w_r[i] *= inv;
  w_i[i] *= inv;
}

// ks (planar out) = s2 (interleaved, from type2) * w (planar)
__global__ void scale_ks_kernel(const float* __restrict__ s2,
                                const float* __restrict__ w_r, const float* __restrict__ w_i,
                                float* __restrict__ k_r, float* __restrict__ k_i) {
  const int i = blockIdx.x * blockDim.x + threadIdx.x;
  if (i >= Bg * Mg) return;
  const int m = i % Mg;
  const float kr = s2[2 * i], ki = s2[2 * i + 1];
  const float wr = w_r[m],    wi = w_i[m];
  k_r[i] = kr * wr - ki * wi;
  k_i[i] = kr * wi + ki * wr;
}

__global__ void abs_kernel(const float* __restrict__ rec_r, const float* __restrict__ rec_i,
                           float* __restrict__ out) {
  const int i = blockIdx.x * blockDim.x + threadIdx.x;
  if (i >= Bg * NXg * NYg) return;
  const float rr = rec_r[i], ri = rec_i[i];
  out[i] = sqrtf(rr * rr + ri * ri);
}

// ---------------------------------------------------------------------------
extern "C" void kernel_launch(void* const* d_in, const int* in_sizes, int n_in,
                              void* d_out, int out_size, void* d_ws, size_t ws_size,
                              hipStream_t stream) {
  (void)in_sizes; (void)n_in; (void)out_size; (void)ws_size;
  const float* image = (const float*)d_in[0];   // [4,256,256] f32 (planar real)
  const float* traj  = (const float*)d_in[1];   // [2,M] f32
  float* out = (float*)d_out;                   // [4,256,256,1] f32

  float* ws = (float*)d_ws;
  v2f* ExC = (v2f*)ws;   ws += (size_t)Mg * NXg;   // M/2 pairs x 256 x v2f
  v2f* ExS = (v2f*)ws;   ws += (size_t)Mg * NXg;
  v2f* EyC = (v2f*)ws;   ws += (size_t)Mg * NYg;
  v2f* EyS = (v2f*)ws;   ws += (size_t)Mg * NYg;
  float* EycP = ws;      ws += (size_t)Mg * NYg;   // planar planes
  float* EysP = ws;      ws += (size_t)Mg * NYg;
  float* w_r = ws;       ws += Mg;
  float* w_i = ws;       ws += Mg;
  float* g_r = ws;       ws += NXg * NYg;          // planar complex grid
  float* g_i = ws;       ws += NXg * NYg;
  float* s  = ws;        ws += 2 * (size_t)Mg;     // interleaved (atomics)
  float* k2 = ws;        ws += 2 * (size_t)Bg * Mg; // type2 batched out (interleaved)
  float* k_r = ws;       ws += (size_t)Bg * Mg;     // planar kspace
  float* k_i = ws;       ws += (size_t)Bg * Mg;
  float* r_r = ws;       ws += (size_t)Bg * NXg * NYg;
  float* r_i = ws;       ws += (size_t)Bg * NXg * NYg;

  build_tables_kernel<<<((Mg / 2) * NXg + 255) / 256, 256, 0, stream>>>(
      traj, ExC, ExS, EyC, EyS, EycP, EysP);
  init_w_kernel<<<(Mg + 255) / 256, 256, 0, stream>>>(w_r, w_i);

  // Pipe-Menon density compensation: w <- w / |type2(type1(w))|
  for (int it = 0; it < 10; ++it) {
    type1_kernel<<<dim3(NXg / 16, NYg / 32, 1), 256, 0, stream>>>(
        ExC, ExS, EyC, EyS, w_r, w_i, g_r, g_i);
    zero_kernel<<<(2 * Mg + 255) / 256, 256, 0, stream>>>(s, 2 * Mg);
    type2_kernel<true><<<dim3(513, NXg / 32, 1), 64, 0, stream>>>(
        (const float*)ExC, (const float*)ExS, (const v2f*)EycP, (const v2f*)EysP,
        (const v2f*)g_r, (const v2f*)g_i, s);
    update_w_kernel<<<(Mg + 255) / 256, 256, 0, stream>>>(w_r, w_i, s);
  }

  // kspace = type2(image) * w   (image is real -> HAS_IMAG=false)
  zero_kernel<<<(2 * Bg * Mg + 255) / 256, 256, 0, stream>>>(k2, 2 * Bg * Mg);
  type2_kernel<false><<<dim3(513, NXg / 32, Bg), 64, 0, stream>>>(
      (const float*)ExC, (const float*)ExS, (const v2f*)EycP, (const v2f*)EysP,
      (const v2f*)image, nullptr, k2);
  scale_ks_kernel<<<(Bg * Mg + 255) / 256, 256, 0, stream>>>(k2, w_r, w_i, k_r, k_i);

  // recon = type1(kspace); out = |recon|
  type1_kernel<<<dim3(NXg / 16, NYg / 32, Bg), 256, 0, stream>>>(
      ExC, ExS, EyC, EyS, k_r, k_i, r_r, r_i);
  abs_kernel<<<(Bg * NXg * NYg + 255) / 256, 256, 0, stream>>>(r_r, r_i, out);
}